// TripletLoss_meta_finetune_5446018531420
// MI455X (gfx1250) — compile-verified
//
#include <hip/hip_runtime.h>
#include <hip/hip_bf16.h>

// Triplet margin loss with hardest-positive / hardest-negative mining.
// n = 8192, labels int32 in [0, 512). Pure HBM-bound row reduction:
// 512 MB mandatory reads -> ~22 us floor at 23.3 TB/s.

#define NN      8192
#define BLOCK   256
#define WAVES   (BLOCK / 32)      // 8 waves per block, one row per wave
#define MARGIN_F 0.3f
#define EPS_F    1e-12f

typedef __attribute__((ext_vector_type(4))) float f4;
typedef __attribute__((ext_vector_type(4))) int   i4;

// The async-to-LDS builtins take GCC-style vector_size(16) int pointers in
// addrspace(1) (global) and addrspace(3) (LDS) — match those types exactly.
typedef int v4i __attribute__((vector_size(16)));
typedef __attribute__((address_space(1))) v4i* g_v4i_ptr;
typedef __attribute__((address_space(3))) v4i* l_v4i_ptr;

#if __has_builtin(__builtin_amdgcn_global_load_async_to_lds_b128) && \
    __has_builtin(__builtin_amdgcn_s_wait_asynccnt)
#define USE_ASYNC_LDS 1
#endif

__global__ __launch_bounds__(BLOCK)
void triplet_rows_kernel(const float* __restrict__ P,   // inputs_p [NN,NN]
                         const float* __restrict__ Q,   // inputs_n [NN,NN]
                         const int*   __restrict__ lab, // targets  [NN]
                         float*       __restrict__ rowLoss) // [NN]
{
    __shared__ __align__(16) int sLab[NN];   // 32 KB label cache per block

    const int t    = threadIdx.x;
    const int wave = t >> 5;
    const int lane = t & 31;
    const int row  = blockIdx.x * WAVES + wave;   // grid = NN/WAVES, always in range

    // ---- Stage all 8192 int32 labels into LDS (32 KB), async if available ----
#ifdef USE_ASYNC_LDS
    #pragma unroll
    for (int k = 0; k < NN / (BLOCK * 4); ++k) {          // 8 iters of b128/lane
        const int j = k * (BLOCK * 4) + t * 4;
        __builtin_amdgcn_global_load_async_to_lds_b128(
            (g_v4i_ptr)(lab + j),
            (l_v4i_ptr)(&sLab[j]),
            /*offset=*/0, /*cpol=*/0);
    }
    __builtin_amdgcn_s_wait_asynccnt(0);                  // this wave's copies done
#else
    #pragma unroll
    for (int k = 0; k < NN / (BLOCK * 4); ++k) {
        const int j = k * (BLOCK * 4) + t * 4;
        *(i4*)(&sLab[j]) = *(const i4*)(lab + j);
    }
#endif
    __syncthreads();                                      // labels visible to all waves

    const float* prow = P + (size_t)row * NN;
    const float* qrow = Q + (size_t)row * NN;
    __builtin_prefetch(prow, 0, 0);                       // global_prefetch_b8
    __builtin_prefetch(qrow, 0, 0);

    const int   labi = sLab[row];
    const float INF  = __builtin_inff();
    float minP =  INF;    // masked min over P row  (hardest positive, pre-sqrt)
    float maxN = -INF;    // plain max over Q row   (hardest negative, pre-sqrt)

    // Each wave streams one full row of each matrix: 64 x b128 per lane per matrix,
    // non-temporal (single-use stream, keep L2 for labels).
    #pragma unroll 4
    for (int k = 0; k < NN / 128; ++k) {
        const int j = k * 128 + lane * 4;
        f4 p = __builtin_nontemporal_load((const f4*)(prow + j));
        f4 q = __builtin_nontemporal_load((const f4*)(qrow + j));
        i4 lb = *(const i4*)(&sLab[j]);

        minP = fminf(minP, (lb.x == labi) ? p.x : INF);
        minP = fminf(minP, (lb.y == labi) ? p.y : INF);
        minP = fminf(minP, (lb.z == labi) ? p.z : INF);
        minP = fminf(minP, (lb.w == labi) ? p.w : INF);

        maxN = fmaxf(maxN, fmaxf(fmaxf(q.x, q.y), fmaxf(q.z, q.w)));
    }

    // Wave32 butterfly reduction (no cross-wave sync needed: one row per wave).
    #pragma unroll
    for (int off = 16; off > 0; off >>= 1) {
        minP = fminf(minP, __shfl_xor(minP, off, 32));
        maxN = fmaxf(maxN, __shfl_xor(maxN, off, 32));
    }

    if (lane == 0) {
        // sqrt/clamp commute with min/max (monotone), so one sqrt per row.
        const float dap = __builtin_sqrtf(fmaxf(minP, EPS_F));
        const float dan = __builtin_sqrtf(fmaxf(maxN, EPS_F));
        rowLoss[row] = fmaxf(dan - dap + MARGIN_F, 0.0f);
    }
}

// Deterministic single-block tree reduction: mean of rowLoss[0..NN) -> out[0].
__global__ __launch_bounds__(256)
void triplet_mean_kernel(const float* __restrict__ rowLoss, float* __restrict__ out)
{
    __shared__ float sSum[8];
    const int t = threadIdx.x;

    float acc = 0.0f;
    #pragma unroll
    for (int k = 0; k < NN / 256; ++k)            // 32 strided adds, fixed order
        acc += rowLoss[k * 256 + t];

    #pragma unroll
    for (int off = 16; off > 0; off >>= 1)
        acc += __shfl_xor(acc, off, 32);

    if ((t & 31) == 0) sSum[t >> 5] = acc;
    __syncthreads();

    if (t == 0) {
        float r = 0.0f;
        #pragma unroll
        for (int w = 0; w < 8; ++w) r += sSum[w];
        out[0] = r * (1.0f / (float)NN);
    }
}

extern "C" void kernel_launch(void* const* d_in, const int* in_sizes, int n_in,
                              void* d_out, int out_size, void* d_ws, size_t ws_size,
                              hipStream_t stream) {
    (void)in_sizes; (void)n_in; (void)out_size; (void)ws_size;

    const float* P   = (const float*)d_in[0];   // inputs_p, fp32 [8192*8192]
    const float* Q   = (const float*)d_in[1];   // inputs_n, fp32 [8192*8192]
    const int*   lab = (const int*)  d_in[2];   // targets_p, int32 [8192]

    float* rowLoss = (float*)d_ws;              // 8192 floats = 32 KB scratch
    float* outF    = (float*)d_out;             // scalar fp32 loss

    triplet_rows_kernel<<<NN / WAVES, BLOCK, 0, stream>>>(P, Q, lab, rowLoss);
    triplet_mean_kernel<<<1, 256, 0, stream>>>(rowLoss, outF);
}